// GCNInceptionLayer_5549097746958
// MI455X (gfx1250) — compile-verified
//
#include <hip/hip_runtime.h>
#include <hip/hip_bf16.h>

#define NPTS   8192
#define FIN    256
#define DD     64
#define KSEL   40
#define KMAXN  9
#define SLOPE  0.01f
#define FINF   3.402823466e38f

typedef __attribute__((ext_vector_type(2))) float v2f;
typedef __attribute__((ext_vector_type(8))) float v8f;

// ---------------------------------------------------------------------------
// CDNA5 fp32 WMMA: D = A(16x4) * B(4x16) + C(16x16)
// A/B frag: lane l, vgpr v -> row l%16, K = 2*(l/16)+v
// C/D frag: lane l, vgpr r -> M = r + 8*(l/16), N = l%16
// ---------------------------------------------------------------------------
__device__ __forceinline__ v8f wmma4(v2f a, v2f b, v8f c) {
  return __builtin_amdgcn_wmma_f32_16x16x4_f32(false, a, false, b,
                                               (short)0, c, false, false);
}

__device__ __forceinline__ v2f gfrag(const float* __restrict__ p, int ld, int r, int k) {
  const float* q = p + (long)r * ld + k;
  v2f v; v.x = q[0]; v.y = q[1];
  return v;
}

__device__ __forceinline__ float leaky_f(float x) { return x >= 0.0f ? x : SLOPE * x; }

// ---------------------------------------------------------------------------
// Kernel 1: f = feature @ Wb.T + bb      [8192,256] x [64,256]^T -> [8192,64]
// ---------------------------------------------------------------------------
__global__ void k_embed(const float* __restrict__ feat, const float* __restrict__ Wb,
                        const float* __restrict__ bb, float* __restrict__ f) {
  const int lane = threadIdx.x & 31, wave = threadIdx.x >> 5;
  const int lm = lane & 15, lhi = lane >> 4;
  const int m0 = (blockIdx.x * 4 + wave) * 16;
  if (m0 >= NPTS) return;

  v8f acc[4];
  v8f zero = {};
  for (int nt = 0; nt < 4; ++nt) acc[nt] = zero;

  for (int k0 = 0; k0 < FIN; k0 += 4) {
    v2f a = gfrag(feat, FIN, m0 + lm, k0 + 2 * lhi);
    for (int nt = 0; nt < 4; ++nt) {
      v2f b = gfrag(Wb, FIN, nt * 16 + lm, k0 + 2 * lhi);
      acc[nt] = wmma4(a, b, acc[nt]);
    }
  }
  for (int nt = 0; nt < 4; ++nt) {
    int col = nt * 16 + lm;
    float bias = bb[col];
    for (int r = 0; r < 8; ++r) {
      int row = m0 + 8 * lhi + r;
      f[row * DD + col] = acc[nt][r] + bias;
    }
  }
}

// ---------------------------------------------------------------------------
// Kernel 2: per-row squared norms
// ---------------------------------------------------------------------------
__global__ void k_fsq(const float* __restrict__ f, float* __restrict__ fsq) {
  int i = blockIdx.x * blockDim.x + threadIdx.x;
  if (i >= NPTS) return;
  float s = 0.0f;
  for (int d = 0; d < DD; ++d) { float v = f[i * DD + d]; s += v * v; }
  fsq[i] = s;
}

// ---------------------------------------------------------------------------
// Kernel 3: pairwise distance + streaming top-40.
// Block = 64 query rows (4 M-tiles per wave), 4 waves -> 64 candidate cols
// per iteration. Each B fragment is reused by 4 WMMAs (4:1 wmma:load),
// cutting L2 candidate traffic 4x vs a 16-row block. fsq[query] is applied
// by the selector thread (constant per row) to save 32 VGPRs here.
// ---------------------------------------------------------------------------
__global__ void k_knn(const float* __restrict__ f, const float* __restrict__ fsq,
                      int* __restrict__ idx40) {
  const int tid = threadIdx.x;
  const int lane = tid & 31, wave = tid >> 5;
  const int lm = lane & 15, lhi = lane >> 4;
  const int q0 = blockIdx.x * 64;

  __shared__ float sDist[64][68];
  __shared__ float sBd[64][KSEL];
  __shared__ int   sBi[64][KSEL];

  // A fragments for 4 query M-tiles (reused across all 128 iterations)
  v2f aFrag[4][16];
  for (int mt = 0; mt < 4; ++mt)
    for (int kk = 0; kk < 16; ++kk)
      aFrag[mt][kk] = gfrag(f, DD, q0 + mt * 16 + lm, kk * 4 + 2 * lhi);

  float fq = 0.0f;
  if (tid < 64) {
    fq = fsq[q0 + tid];
    for (int j = 0; j < KSEL; ++j) { sBd[tid][j] = FINF; sBi[tid][j] = 0; }
  }
  float worst = FINF; int worstPos = 0;
  __syncthreads();

  for (int it = 0; it < NPTS / 64; ++it) {
    const int c0 = it * 64 + wave * 16;
    if (it + 1 < NPTS / 64)
      __builtin_prefetch(&f[((it + 1) * 64 + wave * 16 + lm) * DD], 0, 1);

    v8f acc[4];
    v8f zero = {};
    for (int mt = 0; mt < 4; ++mt) acc[mt] = zero;

    for (int kk = 0; kk < 16; ++kk) {
      v2f b = gfrag(f, DD, c0 + lm, kk * 4 + 2 * lhi);
      for (int mt = 0; mt < 4; ++mt)
        acc[mt] = wmma4(aFrag[mt][kk], b, acc[mt]);
    }
    float fc = fsq[c0 + lm];
    for (int mt = 0; mt < 4; ++mt)
      for (int r = 0; r < 8; ++r)
        sDist[mt * 16 + 8 * lhi + r][wave * 16 + lm] = fc - 2.0f * acc[mt][r];
    __syncthreads();

    if (tid < 64) {
      const int m = tid;
      for (int c = 0; c < 64; ++c) {
        float d = sDist[m][c] + fq;
        if (d < worst) {
          sBd[m][worstPos] = d;
          sBi[m][worstPos] = it * 64 + c;
          float w = sBd[m][0]; int wp = 0;
          for (int j = 1; j < KSEL; ++j) {
            float v = sBd[m][j];
            if (v > w) { w = v; wp = j; }
          }
          worst = w; worstPos = wp;
        }
      }
    }
    __syncthreads();
  }

  // sort ascending by distance (top_k order), emit indices
  if (tid < 64) {
    const int m = tid;
    for (int a = 0; a < KSEL; ++a) {
      int mp = a; float mv = sBd[m][a];
      for (int b2 = a + 1; b2 < KSEL; ++b2)
        if (sBd[m][b2] < mv) { mv = sBd[m][b2]; mp = b2; }
      float td = sBd[m][a]; sBd[m][a] = sBd[m][mp]; sBd[m][mp] = td;
      int ti = sBi[m][a]; sBi[m][a] = sBi[m][mp]; sBi[m][mp] = ti;
      idx40[(q0 + m) * KSEL + a] = sBi[m][a];
    }
  }
}

// ---------------------------------------------------------------------------
// Kernel 4: EdgeConv affine collapse.
//   v = f @ Wt.T ;  u = f @ (Wt+Wp).T + bt + bp   (for both convs)
// ---------------------------------------------------------------------------
__global__ void k_transform(const float* __restrict__ f,
                            const float* __restrict__ Wt1, const float* __restrict__ bt1,
                            const float* __restrict__ Wp1, const float* __restrict__ bp1,
                            const float* __restrict__ Wt2, const float* __restrict__ bt2,
                            const float* __restrict__ Wp2, const float* __restrict__ bp2,
                            float* __restrict__ u1, float* __restrict__ v1,
                            float* __restrict__ u2, float* __restrict__ v2) {
  const int lane = threadIdx.x & 31, wave = threadIdx.x >> 5;
  const int lm = lane & 15, lhi = lane >> 4;
  const int m0 = (blockIdx.x * 4 + wave) * 16;
  if (m0 >= NPTS) return;

  v2f aFrag[16];
  for (int kk = 0; kk < 16; ++kk)
    aFrag[kk] = gfrag(f, DD, m0 + lm, kk * 4 + 2 * lhi);

  v8f zero = {};
  for (int conv = 0; conv < 2; ++conv) {
    const float* Wt = conv ? Wt2 : Wt1;
    const float* Wp = conv ? Wp2 : Wp1;
    const float* bt = conv ? bt2 : bt1;
    const float* bp = conv ? bp2 : bp1;
    float* U = conv ? u2 : u1;
    float* V = conv ? v2 : v1;

    v8f ft[4], ph[4];
    for (int nt = 0; nt < 4; ++nt) { ft[nt] = zero; ph[nt] = zero; }
    for (int kk = 0; kk < 16; ++kk) {
      for (int nt = 0; nt < 4; ++nt) {
        v2f bT = gfrag(Wt, DD, nt * 16 + lm, kk * 4 + 2 * lhi);
        ft[nt] = wmma4(aFrag[kk], bT, ft[nt]);
        v2f bP = gfrag(Wp, DD, nt * 16 + lm, kk * 4 + 2 * lhi);
        ph[nt] = wmma4(aFrag[kk], bP, ph[nt]);
      }
    }
    for (int nt = 0; nt < 4; ++nt) {
      int col = nt * 16 + lm;
      float bias = bt[col] + bp[col];
      for (int r = 0; r < 8; ++r) {
        int row = m0 + 8 * lhi + r;
        V[row * DD + col] = ft[nt][r];
        U[row * DD + col] = ft[nt][r] + ph[nt][r] + bias;
      }
    }
  }
}

// ---------------------------------------------------------------------------
// Kernel 5: EdgeConv MLP. 4 nodes/block * 20 neighbors = 80 edge rows.
//   E = leaky(u_i - v_j) ;  Y = leaky(E@W1.T + b1) ;  Z = Y@W2.T + b2 ;
//   h = max_j Z -> hcat[:, colOff:colOff+64]
// ---------------------------------------------------------------------------
#define EC_NODES 4
#define EC_ROWS  (EC_NODES * 20)   // 80 -> 5 M-tiles
#define EC_PAD   68

__global__ void k_edgeconv(const float* __restrict__ u, const float* __restrict__ v,
                           const int* __restrict__ idx40, int step,
                           const float* __restrict__ W1, const float* __restrict__ b1,
                           const float* __restrict__ W2, const float* __restrict__ b2,
                           float* __restrict__ hout /* hcat + col offset, ld=192 */) {
  const int tid = threadIdx.x;
  const int lane = tid & 31, wave = tid >> 5;
  const int lm = lane & 15, lhi = lane >> 4;
  const int node0 = blockIdx.x * EC_NODES;

  __shared__ float sE[EC_ROWS][EC_PAD];
  __shared__ float sY[EC_ROWS][EC_PAD];

  // Build E = leaky(u_i - v_j)
  for (int e = tid; e < EC_ROWS * DD; e += blockDim.x) {
    int r = e >> 6, d = e & 63;
    int n = r / 20, j = r - n * 20;
    int nbr = idx40[(node0 + n) * KSEL + j * step];
    sE[r][d] = leaky_f(u[(node0 + n) * DD + d] - v[nbr * DD + d]);
  }
  __syncthreads();

  // GEMM1: Y = leaky(E @ W1.T + b1)   (each wave owns one N-tile, 5 M-tiles)
  {
    const int nt = wave;
    for (int mt = 0; mt < 5; ++mt) {
      v8f acc = {};
      for (int kk = 0; kk < 16; ++kk) {
        int ar = mt * 16 + lm, ak = kk * 4 + 2 * lhi;
        v2f a; a.x = sE[ar][ak]; a.y = sE[ar][ak + 1];
        v2f b = gfrag(W1, DD, nt * 16 + lm, ak);
        acc = wmma4(a, b, acc);
      }
      int col = nt * 16 + lm;
      float bias = b1[col];
      for (int r = 0; r < 8; ++r)
        sY[mt * 16 + 8 * lhi + r][col] = leaky_f(acc[r] + bias);
    }
  }
  __syncthreads();

  // GEMM2: Z = Y @ W2.T + b2  (into sE)
  {
    const int nt = wave;
    for (int mt = 0; mt < 5; ++mt) {
      v8f acc = {};
      for (int kk = 0; kk < 16; ++kk) {
        int ar = mt * 16 + lm, ak = kk * 4 + 2 * lhi;
        v2f a; a.x = sY[ar][ak]; a.y = sY[ar][ak + 1];
        v2f b = gfrag(W2, DD, nt * 16 + lm, ak);
        acc = wmma4(a, b, acc);
      }
      int col = nt * 16 + lm;
      float bias = b2[col];
      for (int r = 0; r < 8; ++r)
        sE[mt * 16 + 8 * lhi + r][col] = acc[r] + bias;
    }
  }
  __syncthreads();

  // Max-reduce over 20 neighbors
  for (int e = tid; e < EC_NODES * DD; e += blockDim.x) {
    int n = e >> 6, d = e & 63;
    float mx = -FINF;
    for (int j = 0; j < 20; ++j) mx = fmaxf(mx, sE[n * 20 + j][d]);
    hout[(node0 + n) * 192 + d] = mx;
  }
}

// ---------------------------------------------------------------------------
// Kernel 6: h_max = max over 9-NN of f -> hcat[:, 128:192]
// ---------------------------------------------------------------------------
__global__ void k_hmax(const float* __restrict__ f, const int* __restrict__ idx40,
                       float* __restrict__ hcat) {
  int e = blockIdx.x * blockDim.x + threadIdx.x;
  if (e >= NPTS * DD) return;
  int n = e >> 6, d = e & 63;
  float mx = -FINF;
  for (int j = 0; j < KMAXN; ++j) {
    int idx = idx40[n * KSEL + j];
    mx = fmaxf(mx, f[idx * DD + d]);
  }
  hcat[n * 192 + 128 + d] = mx;
}

// ---------------------------------------------------------------------------
// Kernel 7: out = hcat @ Wl.T + bl + feature   [8192,192]x[256,192]^T
// ---------------------------------------------------------------------------
__global__ void k_final(const float* __restrict__ hcat, const float* __restrict__ Wl,
                        const float* __restrict__ bl, const float* __restrict__ feat,
                        float* __restrict__ out) {
  const int lane = threadIdx.x & 31, wave = threadIdx.x >> 5;
  const int lm = lane & 15, lhi = lane >> 4;
  const int m0 = (blockIdx.x * 4 + wave) * 16;
  if (m0 >= NPTS) return;

  v8f acc[16];
  v8f zero = {};
  for (int nt = 0; nt < 16; ++nt) acc[nt] = zero;

  for (int k0 = 0; k0 < 192; k0 += 4) {
    v2f a = gfrag(hcat, 192, m0 + lm, k0 + 2 * lhi);
    for (int nt = 0; nt < 16; ++nt) {
      v2f b = gfrag(Wl, 192, nt * 16 + lm, k0 + 2 * lhi);
      acc[nt] = wmma4(a, b, acc[nt]);
    }
  }
  for (int nt = 0; nt < 16; ++nt) {
    int col = nt * 16 + lm;
    float bias = bl[col];
    for (int r = 0; r < 8; ++r) {
      int row = m0 + 8 * lhi + r;
      out[row * FIN + col] = acc[nt][r] + bias + feat[row * FIN + col];
    }
  }
}

// ---------------------------------------------------------------------------
// Host launch
// ---------------------------------------------------------------------------
extern "C" void kernel_launch(void* const* d_in, const int* in_sizes, int n_in,
                              void* d_out, int out_size, void* d_ws, size_t ws_size,
                              hipStream_t stream) {
  const float* feature = (const float*)d_in[0];
  const float* Wb  = (const float*)d_in[1];
  const float* bb  = (const float*)d_in[2];
  const float* Wt1 = (const float*)d_in[3];
  const float* bt1 = (const float*)d_in[4];
  const float* Wp1 = (const float*)d_in[5];
  const float* bp1 = (const float*)d_in[6];
  const float* Wm1a = (const float*)d_in[7];
  const float* bm1a = (const float*)d_in[8];
  const float* Wm1b = (const float*)d_in[9];
  const float* bm1b = (const float*)d_in[10];
  const float* Wt2 = (const float*)d_in[11];
  const float* bt2 = (const float*)d_in[12];
  const float* Wp2 = (const float*)d_in[13];
  const float* bp2 = (const float*)d_in[14];
  const float* Wm2a = (const float*)d_in[15];
  const float* bm2a = (const float*)d_in[16];
  const float* Wm2b = (const float*)d_in[17];
  const float* bm2b = (const float*)d_in[18];
  const float* Wl = (const float*)d_in[19];
  const float* bl = (const float*)d_in[20];
  float* out = (float*)d_out;

  // Workspace layout (~17.3 MB)
  float* f    = (float*)d_ws;            // 8192*64
  float* fsq  = f + NPTS * DD;           // 8192
  float* u1   = fsq + NPTS;              // 8192*64
  float* v1   = u1 + NPTS * DD;
  float* u2   = v1 + NPTS * DD;
  float* v2   = u2 + NPTS * DD;
  float* hcat = v2 + NPTS * DD;          // 8192*192
  int* idx40  = (int*)(hcat + NPTS * 192); // 8192*40

  k_embed<<<NPTS / 64, 128, 0, stream>>>(feature, Wb, bb, f);
  k_fsq<<<NPTS / 256, 256, 0, stream>>>(f, fsq);
  k_knn<<<NPTS / 64, 128, 0, stream>>>(f, fsq, idx40);
  k_transform<<<NPTS / 64, 128, 0, stream>>>(f, Wt1, bt1, Wp1, bp1,
                                             Wt2, bt2, Wp2, bp2, u1, v1, u2, v2);
  // conv1: idx20 = idx40[:, :20] (step 1) -> hcat cols [0,64)
  k_edgeconv<<<NPTS / EC_NODES, 128, 0, stream>>>(u1, v1, idx40, 1,
                                                  Wm1a, bm1a, Wm1b, bm1b, hcat);
  // conv2: idx20_d2 = idx40[:, ::2] (step 2) -> hcat cols [64,128)
  k_edgeconv<<<NPTS / EC_NODES, 128, 0, stream>>>(u2, v2, idx40, 2,
                                                  Wm2a, bm2a, Wm2b, bm2b, hcat + 64);
  k_hmax<<<NPTS * DD / 256, 256, 0, stream>>>(f, idx40, hcat);
  k_final<<<NPTS / 64, 128, 0, stream>>>(hcat, Wl, bl, feature, out);
}